// SoftPool2d_850403524762
// MI455X (gfx1250) — compile-verified
//
#include <hip/hip_runtime.h>
#include <cstdint>

// SoftPool2d 2x2/s2 on (16,96,256,256) f32 -> (16,96,128,128) f32.
// Pure-bandwidth kernel: ~503 MB streamed => ~21.6us floor at 23.3 TB/s.
// Data path: async global->LDS (b128, NT) stage of a contiguous 16KB tile
// (16 input rows), then LDS reads + 4x softpool + NT float4 store.

typedef float f32x4 __attribute__((ext_vector_type(4)));

#define BC_TOTAL   1536      // B*C = 16*96
#define W_IN       256
#define H_IN       256
#define W_OUT      128
#define H_OUT      128
#define IN_ROWS    16        // input rows staged per block
#define OUT_ROWS   8         // output rows produced per block
#define TILE_FLTS  (IN_ROWS * W_IN)   // 4096 floats = 16 KB
#define THREADS    256

__device__ __forceinline__ float softpool4(float t0, float t1, float t2, float t3) {
    float m  = fmaxf(fmaxf(t0, t1), fmaxf(t2, t3));
    float e0 = __expf(t0 - m);
    float e1 = __expf(t1 - m);
    float e2 = __expf(t2 - m);
    float e3 = __expf(t3 - m);
    float num = t0 * e0 + t1 * e1 + t2 * e2 + t3 * e3;
    float den = e0 + e1 + e2 + e3;
    return __fdividef(num, den);
}

__global__ __launch_bounds__(THREADS) void softpool2x2_kernel(
        const float* __restrict__ x, float* __restrict__ out) {
    __shared__ f32x4 tile4[TILE_FLTS / 4];   // 16 KB, 16B-aligned

    const uint32_t t      = threadIdx.x;
    const uint32_t blk    = blockIdx.x;
    const uint32_t bc     = blk >> 4;        // image index (0..1535)
    const uint32_t rowBlk = blk & 15u;       // 16-row band of the image

    // ---- async stage: 16 KB contiguous input band -> LDS ----------------
    const uint64_t gbase = (uint64_t)(uintptr_t)x +
        ((uint64_t)bc * (uint64_t)(H_IN * W_IN) +
         (uint64_t)rowBlk * (uint64_t)TILE_FLTS) * 4ull;
    const uint32_t lbase = (uint32_t)(uintptr_t)(&tile4[0]);

#pragma unroll
    for (uint32_t k = 0; k < 4; ++k) {
        uint64_t ga = gbase + (uint64_t)(t * 16u + k * 4096u);
        uint32_t la = lbase + t * 16u + k * 4096u;
        // streaming load, bypassing VGPRs; tracked by ASYNCcnt
        asm volatile("global_load_async_to_lds_b128 %0, %1, off th:TH_LOAD_NT"
                     :: "v"(la), "v"(ga) : "memory");
    }
#if __has_builtin(__builtin_amdgcn_s_wait_asynccnt)
    __builtin_amdgcn_s_wait_asynccnt(0);
#else
    asm volatile("s_wait_asynccnt 0" ::: "memory");
#endif
    __syncthreads();

    // ---- compute: each thread -> 4 horizontally-adjacent outputs --------
    const uint32_t r = t >> 5;    // local output row 0..7
    const uint32_t q = t & 31u;   // output quad 0..31 (covers cols 4q..4q+3)

    const f32x4* trow0 = &tile4[(2u * r)      * (W_IN / 4)];
    const f32x4* trow1 = &tile4[(2u * r + 1u) * (W_IN / 4)];
    f32x4 a0 = trow0[2u * q];       // top row, cols 8q..8q+3
    f32x4 a1 = trow0[2u * q + 1u];  // top row, cols 8q+4..8q+7
    f32x4 b0 = trow1[2u * q];       // bottom row
    f32x4 b1 = trow1[2u * q + 1u];

    f32x4 res;
    res.x = softpool4(a0.x, a0.y, b0.x, b0.y);
    res.y = softpool4(a0.z, a0.w, b0.z, b0.w);
    res.z = softpool4(a1.x, a1.y, b1.x, b1.y);
    res.w = softpool4(a1.z, a1.w, b1.z, b1.w);

    const uint64_t ooff = (uint64_t)bc * (uint64_t)(H_OUT * W_OUT) +
                          (uint64_t)(rowBlk * OUT_ROWS + r) * (uint64_t)W_OUT +
                          (uint64_t)(4u * q);
    __builtin_nontemporal_store(res, (f32x4*)(out + ooff));
}

extern "C" void kernel_launch(void* const* d_in, const int* in_sizes, int n_in,
                              void* d_out, int out_size, void* d_ws, size_t ws_size,
                              hipStream_t stream) {
    (void)in_sizes; (void)n_in; (void)out_size; (void)d_ws; (void)ws_size;
    const float* x  = (const float*)d_in[0];
    float*      out = (float*)d_out;
    dim3 grid(BC_TOTAL * 16);   // 24,576 blocks
    dim3 block(THREADS);
    softpool2x2_kernel<<<grid, block, 0, stream>>>(x, out);
}